// LuongAttnDecoderRNN_48180943127079
// MI455X (gfx1250) — compile-verified
//
#include <hip/hip_runtime.h>
#include <math.h>

typedef __attribute__((ext_vector_type(2))) float v2f;
typedef __attribute__((ext_vector_type(4))) float v4f;
typedef __attribute__((ext_vector_type(8))) float v8f;

constexpr int kV = 32000;
constexpr int kH = 1024;
constexpr int kB = 64;
constexpr int kT = 1024;
constexpr int kSplit = 8;            // T-dim split for attention
constexpr int kChunk = kT / kSplit;  // rows per attention block

// ---------------------------------------------------------------------------
// Generic FP32 WMMA GEMM:  C[64, N] = A[64, K] @ W[N, K]^T + bias  (opt tanh)
// K is a compile-time constant so the 4 A-row bases fold into immediate
// offsets and the K-loop advances a single base pointer. The K-loop is
// explicitly double-buffered: chunk k+4 is loaded while chunk k feeds the
// 4 WMMAs, so the s_wait before each WMMA group covers loads issued a full
// iteration (x unroll window) earlier instead of the immediately preceding
// ones.
// f32 A 16x4 fragment: lanes 0-15 hold (k,k+1) of row M=lane; lanes 16-31
// hold (k+2,k+3) of row M=lane-16  -> one contiguous 8-byte load per lane.
// B (=W^T) 4x16 fragment mirrors this. C/D v8f: VGPR r -> row r+half*8.
// ---------------------------------------------------------------------------
template <int ACT, int K>
__global__ __launch_bounds__(256) void gemm_f32_wmma(
    const float* __restrict__ A, const float* __restrict__ W,
    const float* __restrict__ bias, float* __restrict__ C, int N) {
  const int lane = threadIdx.x & 31;
  const int wave = threadIdx.x >> 5;
  const int nt   = blockIdx.x * 8 + wave;   // 16-wide N tile index
  const int l16  = lane & 15;
  const int half = lane >> 4;
  const int n_col = nt * 16 + l16;

  const float* wrow = W + (size_t)n_col * K + half * 2;
  const float* arow = A + (size_t)l16 * K + half * 2;

  v8f acc0 = {}, acc1 = {}, acc2 = {}, acc3 = {};

  // ---- pipeline prologue: preload chunk 0 -------------------------------
  v2f bc  = *(const v2f*)(wrow);
  v2f a0c = *(const v2f*)(arow);
  v2f a1c = *(const v2f*)(arow + 16 * K);
  v2f a2c = *(const v2f*)(arow + 32 * K);
  v2f a3c = *(const v2f*)(arow + 48 * K);

#pragma unroll 8
  for (int k = 4; k < K; k += 4) {
    // loads for the NEXT chunk issue before the WMMAs consuming the current
    v2f bn  = *(const v2f*)(wrow + k);
    v2f a0n = *(const v2f*)(arow + k);
    v2f a1n = *(const v2f*)(arow + 16 * K + k);
    v2f a2n = *(const v2f*)(arow + 32 * K + k);
    v2f a3n = *(const v2f*)(arow + 48 * K + k);
    if ((k & 31) == 4)  // once per 32-float chunk: pull W ~2KB ahead
      __builtin_prefetch(wrow + k + 512, 0, 1);
    acc0 = __builtin_amdgcn_wmma_f32_16x16x4_f32(false, a0c, false, bc,
                                                 (short)0, acc0, false, false);
    acc1 = __builtin_amdgcn_wmma_f32_16x16x4_f32(false, a1c, false, bc,
                                                 (short)0, acc1, false, false);
    acc2 = __builtin_amdgcn_wmma_f32_16x16x4_f32(false, a2c, false, bc,
                                                 (short)0, acc2, false, false);
    acc3 = __builtin_amdgcn_wmma_f32_16x16x4_f32(false, a3c, false, bc,
                                                 (short)0, acc3, false, false);
    bc = bn; a0c = a0n; a1c = a1n; a2c = a2n; a3c = a3n;
  }
  // ---- pipeline epilogue: last chunk ------------------------------------
  acc0 = __builtin_amdgcn_wmma_f32_16x16x4_f32(false, a0c, false, bc,
                                               (short)0, acc0, false, false);
  acc1 = __builtin_amdgcn_wmma_f32_16x16x4_f32(false, a1c, false, bc,
                                               (short)0, acc1, false, false);
  acc2 = __builtin_amdgcn_wmma_f32_16x16x4_f32(false, a2c, false, bc,
                                               (short)0, acc2, false, false);
  acc3 = __builtin_amdgcn_wmma_f32_16x16x4_f32(false, a3c, false, bc,
                                               (short)0, acc3, false, false);

  const float bv = bias[n_col];
#pragma unroll
  for (int r = 0; r < 8; ++r) {
    const int row = r + half * 8;
    float v0 = acc0[r] + bv;
    float v1 = acc1[r] + bv;
    float v2 = acc2[r] + bv;
    float v3 = acc3[r] + bv;
    if (ACT == 1) { v0 = tanhf(v0); v1 = tanhf(v1); v2 = tanhf(v2); v3 = tanhf(v3); }
    C[(size_t)(row)      * N + n_col] = v0;
    C[(size_t)(row + 16) * N + n_col] = v1;
    C[(size_t)(row + 32) * N + n_col] = v2;
    C[(size_t)(row + 48) * N + n_col] = v3;
  }
}

// ---------------------------------------------------------------------------
// Embedding gather: embedded[b, h] = emb_table[seq[b], h]
// ---------------------------------------------------------------------------
__global__ __launch_bounds__(256) void embed_gather(
    const int* __restrict__ seq, const float* __restrict__ emb,
    float* __restrict__ out) {
  const int idx = blockIdx.x * 256 + threadIdx.x;      // B*H threads
  const int b = idx >> 10, h = idx & (kH - 1);
  out[idx] = emb[(size_t)seq[b] * kH + h];
}

// ---------------------------------------------------------------------------
// GRU elementwise (biases already folded into gx/gh by the GEMM epilogues)
// ---------------------------------------------------------------------------
__device__ __forceinline__ float sigmoidf(float x) {
  return 1.0f / (1.0f + __expf(-x));
}

__global__ __launch_bounds__(256) void gru_elem(
    const float* __restrict__ gx, const float* __restrict__ gh,
    const float* __restrict__ hprev, float* __restrict__ h_new,
    float* __restrict__ hid_out, float* __restrict__ concat_in) {
  const int idx = blockIdx.x * 256 + threadIdx.x;      // B*H threads
  const int b = idx >> 10, j = idx & (kH - 1);
  const float* gxb = gx + (size_t)b * 3 * kH;
  const float* ghb = gh + (size_t)b * 3 * kH;
  const float r = sigmoidf(gxb[j] + ghb[j]);
  const float z = sigmoidf(gxb[kH + j] + ghb[kH + j]);
  const float n = tanhf(gxb[2 * kH + j] + r * ghb[2 * kH + j]);
  const float hv = (1.0f - z) * n + z * hprev[idx];
  h_new[idx] = hv;
  hid_out[idx] = hv;
  concat_in[(size_t)b * 2 * kH + j] = hv;
}

// ---------------------------------------------------------------------------
// Fused attention, pass 1: online softmax over a 128-row slice of T.
// Grid = kB * kSplit blocks (512) for enough memory-level parallelism to
// stream the 268 MB encoder tensor at HBM rate. Each block emits a partial
// (M, L, unnormalized ctx) plus the raw scores for the attn_weights output.
// ---------------------------------------------------------------------------
__global__ __launch_bounds__(256) void attn_partial(
    const float* __restrict__ h_new, const float* __restrict__ enc,
    float* __restrict__ scores, float* __restrict__ pm,
    float* __restrict__ pl, float* __restrict__ pctx) {
  const int blk = blockIdx.x;           // kB * kSplit
  const int b = blk >> 3;               // kSplit == 8
  const int s = blk & 7;
  const int tid = threadIdx.x;
  const int lane = tid & 31;
  const int wave = tid >> 5;

  __shared__ float lds_m[8];
  __shared__ float lds_l[8];
  __shared__ float lds_ctx[kH];
  __shared__ float lds_lsum;

  for (int i = tid; i < kH; i += 256) lds_ctx[i] = 0.0f;
  if (tid == 0) lds_lsum = 0.0f;
  __syncthreads();

  // Cache this batch's query row: lane owns h = c*128 + lane*4 .. +3
  v4f hv[8];
#pragma unroll
  for (int c = 0; c < 8; ++c)
    hv[c] = *(const v4f*)(h_new + (size_t)b * kH + c * 128 + lane * 4);

  v4f ctx[8];
#pragma unroll
  for (int c = 0; c < 8; ++c) ctx[c] = (v4f){0.f, 0.f, 0.f, 0.f};
  float m = -__builtin_inff();
  float l = 0.0f;

  const int t0 = s * kChunk;
  for (int t = t0 + wave; t < t0 + kChunk; t += 8) {
    const float* row = enc + ((size_t)t * kB + b) * kH;
    v4f ev[8];
    float sc = 0.0f;
#pragma unroll
    for (int c = 0; c < 8; ++c) {
      ev[c] = *(const v4f*)(row + c * 128 + lane * 4);
      sc += ev[c].x * hv[c].x + ev[c].y * hv[c].y +
            ev[c].z * hv[c].z + ev[c].w * hv[c].w;
    }
#pragma unroll
    for (int off = 16; off > 0; off >>= 1) sc += __shfl_xor(sc, off, 32);
    if (lane == 0) scores[(size_t)b * kT + t] = sc;

    const float m_new = fmaxf(m, sc);
    const float scale = __expf(m - m_new);
    const float p = __expf(sc - m_new);
    l = l * scale + p;
#pragma unroll
    for (int c = 0; c < 8; ++c) ctx[c] = ctx[c] * scale + ev[c] * p;
    m = m_new;
  }

  if (lane == 0) { lds_m[wave] = m; lds_l[wave] = l; }
  __syncthreads();

  float M = -__builtin_inff();
#pragma unroll
  for (int w = 0; w < 8; ++w) M = fmaxf(M, lds_m[w]);
  const float f = __expf(m - M);
  if (lane == 0) atomicAdd(&lds_lsum, l * f);
#pragma unroll
  for (int c = 0; c < 8; ++c) {
    const int base = c * 128 + lane * 4;
    atomicAdd(&lds_ctx[base + 0], ctx[c].x * f);
    atomicAdd(&lds_ctx[base + 1], ctx[c].y * f);
    atomicAdd(&lds_ctx[base + 2], ctx[c].z * f);
    atomicAdd(&lds_ctx[base + 3], ctx[c].w * f);
  }
  __syncthreads();

  for (int i = tid; i < kH; i += 256)
    pctx[(size_t)blk * kH + i] = lds_ctx[i];
  if (tid == 0) { pm[blk] = M; pl[blk] = lds_lsum; }
}

// ---------------------------------------------------------------------------
// Attention pass 2: merge the kSplit partials per batch element.
// ---------------------------------------------------------------------------
__global__ __launch_bounds__(256) void attn_reduce(
    const float* __restrict__ pm, const float* __restrict__ pl,
    const float* __restrict__ pctx, float* __restrict__ m_l,
    float* __restrict__ concat_in) {
  const int b = blockIdx.x;
  const int tid = threadIdx.x;

  float M = -__builtin_inff();
#pragma unroll
  for (int s = 0; s < kSplit; ++s) M = fmaxf(M, pm[b * kSplit + s]);
  float f[kSplit];
  float L = 0.0f;
#pragma unroll
  for (int s = 0; s < kSplit; ++s) {
    f[s] = __expf(pm[b * kSplit + s] - M);
    L += pl[b * kSplit + s] * f[s];
  }
  for (int i = tid; i < kH; i += 256) {
    float c = 0.0f;
#pragma unroll
    for (int s = 0; s < kSplit; ++s)
      c += pctx[(size_t)(b * kSplit + s) * kH + i] * f[s];
    concat_in[(size_t)b * 2 * kH + kH + i] = c / L;
  }
  if (tid == 0) { m_l[2 * b] = M; m_l[2 * b + 1] = L; }
}

__global__ __launch_bounds__(256) void attn_norm(
    const float* __restrict__ scores, const float* __restrict__ m_l,
    float* __restrict__ attn_out) {
  const int idx = blockIdx.x * 256 + threadIdx.x;      // B*T threads
  const int b = idx >> 10;
  attn_out[idx] = __expf(scores[idx] - m_l[2 * b]) / m_l[2 * b + 1];
}

// ---------------------------------------------------------------------------
extern "C" void kernel_launch(void* const* d_in, const int* in_sizes, int n_in,
                              void* d_out, int out_size, void* d_ws,
                              size_t ws_size, hipStream_t stream) {
  const int*   seq   = (const int*)d_in[0];
  const float* hprev = (const float*)d_in[1];   // (1, B, H)
  const float* enc   = (const float*)d_in[2];   // (T, B, H)
  const float* emb   = (const float*)d_in[3];   // (V, H)
  const float* w_ih  = (const float*)d_in[4];   // (3H, H)
  const float* w_hh  = (const float*)d_in[5];   // (3H, H)
  const float* b_ih  = (const float*)d_in[6];
  const float* b_hh  = (const float*)d_in[7];
  const float* cW    = (const float*)d_in[8];   // (H, 2H)
  const float* cb    = (const float*)d_in[9];
  const float* oW    = (const float*)d_in[10];  // (V, H)
  const float* ob    = (const float*)d_in[11];

  float* out      = (float*)d_out;              // (B, V)
  float* hid_out  = out + (size_t)kB * kV;      // (1, B, H)
  float* attn_out = hid_out + (size_t)kB * kH;  // (B, 1, T)

  float* ws         = (float*)d_ws;
  float* embedded   = ws;                                // B*H
  float* gx         = embedded + kB * kH;                // B*3H
  float* gh         = gx + kB * 3 * kH;                  // B*3H
  float* h_new      = gh + kB * 3 * kH;                  // B*H
  float* scores     = h_new + kB * kH;                   // B*T
  float* m_l        = scores + kB * kT;                  // 2*B
  float* concat_in  = m_l + 2 * kB;                      // B*2H
  float* concat_out = concat_in + kB * 2 * kH;           // B*H
  float* pm         = concat_out + kB * kH;              // B*kSplit
  float* pl         = pm + kB * kSplit;                  // B*kSplit
  float* pctx       = pl + kB * kSplit;                  // B*kSplit*H

  embed_gather<<<(kB * kH) / 256, 256, 0, stream>>>(seq, emb, embedded);

  // GRU gate GEMMs (N = 3H = 3072 -> 192 tiles -> 24 blocks), bias folded in.
  gemm_f32_wmma<0, kH><<<24, 256, 0, stream>>>(embedded, w_ih, b_ih, gx, 3 * kH);
  gemm_f32_wmma<0, kH><<<24, 256, 0, stream>>>(hprev,    w_hh, b_hh, gh, 3 * kH);

  gru_elem<<<(kB * kH) / 256, 256, 0, stream>>>(gx, gh, hprev, h_new, hid_out,
                                                concat_in);

  attn_partial<<<kB * kSplit, 256, 0, stream>>>(h_new, enc, scores, pm, pl,
                                                pctx);
  attn_reduce<<<kB, 256, 0, stream>>>(pm, pl, pctx, m_l, concat_in);
  attn_norm<<<(kB * kT) / 256, 256, 0, stream>>>(scores, m_l, attn_out);

  // concat GEMM with tanh (N = H = 1024 -> 64 tiles -> 8 blocks), K = 2H.
  gemm_f32_wmma<1, 2 * kH><<<8, 256, 0, stream>>>(concat_in, cW, cb, concat_out,
                                                  kH);

  // Vocab projection (N = 32000 -> 2000 tiles -> 250 blocks).
  gemm_f32_wmma<0, kH><<<250, 256, 0, stream>>>(concat_out, oW, ob, out, kV);
}